// QLayer_16166256902301
// MI455X (gfx1250) — compile-verified
//
#include <hip/hip_runtime.h>
#include <math.h>

#define DIM 512
#define BATCH 32
#define BM 128
#define BN 64
#define BK 32
#define BKP 34              // padded LDS K-stride (even -> float2 aligned, conflict-free)
#define SQUARINGS 8
#define TAYLOR_ORDER 8
#define INV2S (1.0f/256.0f) // 1 / 2^SQUARINGS

typedef float v2f __attribute__((ext_vector_type(2)));
typedef float v8f __attribute__((ext_vector_type(8)));
typedef int   v2i __attribute__((ext_vector_type(2)));

// gfx1250 async global->LDS staging (ASYNCcnt-tracked).  Signatures from the
// compiler: b64 = (AS1 int2* src, AS3 int2* dst, imm off, imm cpol),
//           b32 = (AS1 int*  src, AS3 int*  dst, imm off, imm cpol).
#if defined(__AMDGCN__) && \
    __has_builtin(__builtin_amdgcn_global_load_async_to_lds_b64) && \
    __has_builtin(__builtin_amdgcn_global_load_async_to_lds_b32) && \
    __has_builtin(__builtin_amdgcn_s_wait_asynccnt)
#define USE_ASYNC_LDS 1
#define ASG64(p) ((__attribute__((address_space(1))) v2i*)(p))
#define ASL64(p) ((__attribute__((address_space(3))) v2i*)(p))
#define ASG32(p) ((__attribute__((address_space(1))) int*)(p))
#define ASL32(p) ((__attribute__((address_space(3))) int*)(p))
#else
#define USE_ASYNC_LDS 0
#endif

// ---------------------------------------------------------------------------
// Kernel 1: per-index tables.
//   msup[j] = Ms[j-1][j] = -conj(alpha)*sqrt(j)  / 2^s   (0 for j==0)
//   msub[j] = Ms[j+1][j] =  alpha      *sqrt(j+1)/ 2^s   (0 for j==DIM-1)
//   dR[n] = exp(-i (th1+th2) n),  dK[n] = exp(+i kappa n^2)  (double-reduced phase)
// ---------------------------------------------------------------------------
__global__ void qod_tables(const float* th1, const float* th2,
                           const float* bxp, const float* bpp, const float* kapp,
                           float2* msup, float2* msub, float2* dR, float2* dK) {
    int n = threadIdx.x;
    float ax = bxp[0], ap = bpp[0];

    float sj = sqrtf((float)n);
    float2 sup = make_float2(-ax * sj * INV2S, ap * sj * INV2S);
    if (n == 0) sup = make_float2(0.f, 0.f);
    msup[n] = sup;

    float sj1 = sqrtf((float)(n + 1));
    float2 sub = make_float2(ax * sj1 * INV2S, ap * sj1 * INV2S);
    if (n == DIM - 1) sub = make_float2(0.f, 0.f);
    msub[n] = sub;

    const double TWO_PI = 6.283185307179586476925286766559;
    double th = (double)th1[0] + (double)th2[0];
    double phR = fmod(-th * (double)n, TWO_PI);
    dR[n] = make_float2((float)cos(phR), (float)sin(phR));
    double phK = fmod((double)kapp[0] * (double)n * (double)n, TWO_PI);
    dK[n] = make_float2((float)cos(phK), (float)sin(phK));
}

// ---------------------------------------------------------------------------
// Kernel 2: E = I + Ms ; P = Ms   (dense planes, Ms tridiagonal w/ zero diag)
// ---------------------------------------------------------------------------
__global__ void qod_taylor_init(float* Ere, float* Eim, float* Pre, float* Pim,
                                const float2* msup, const float2* msub) {
    int idx = blockIdx.x * blockDim.x + threadIdx.x;
    int i = idx >> 9, j = idx & (DIM - 1);
    float mr = 0.f, mi = 0.f;
    if (j == i + 1)      { mr = msup[j].x; mi = msup[j].y; }
    else if (j + 1 == i) { mr = msub[j].x; mi = msub[j].y; }
    Pre[idx] = mr;
    Pim[idx] = mi;
    Ere[idx] = mr + ((i == j) ? 1.f : 0.f);
    Eim[idx] = mi;
}

// ---------------------------------------------------------------------------
// Kernel 3: one Taylor term.  Q = (P @ Ms)/k  (Ms tridiagonal -> 2-point stencil
// along the row), E += Q.  O(dim^2) per term.
// ---------------------------------------------------------------------------
__global__ void qod_taylor_step(float* Ere, float* Eim,
                                const float* Pre, const float* Pim,
                                float* Qre, float* Qim,
                                const float2* msup, const float2* msub, float invk) {
    int idx = blockIdx.x * blockDim.x + threadIdx.x;
    int j = idx & (DIM - 1);
    float lr = 0.f, li = 0.f;
    if (j > 0) {
        float pr = Pre[idx - 1], pi = Pim[idx - 1];
        float2 m = msup[j];
        lr += pr * m.x - pi * m.y;
        li += pr * m.y + pi * m.x;
    }
    if (j < DIM - 1) {
        float pr = Pre[idx + 1], pi = Pim[idx + 1];
        float2 m = msub[j];
        lr += pr * m.x - pi * m.y;
        li += pr * m.y + pi * m.x;
    }
    lr *= invk; li *= invk;
    Qre[idx] = lr; Qim[idx] = li;
    Ere[idx] += lr; Eim[idx] += li;
}

// ---------------------------------------------------------------------------
// Kernel 4: fuse diagonals: W[i][j] = dK[i] * E[i][j] * dR[j]
// ---------------------------------------------------------------------------
__global__ void qod_fuse(const float* Ere, const float* Eim,
                         const float2* dK, const float2* dR,
                         float* Wre, float* Wim) {
    int idx = blockIdx.x * blockDim.x + threadIdx.x;
    int i = idx >> 9, j = idx & (DIM - 1);
    float2 a = dK[i], b = dR[j];
    float sr = a.x * b.x - a.y * b.y;
    float si = a.x * b.y + a.y * b.x;
    float er = Ere[idx], ei = Eim[idx];
    Wre[idx] = er * sr - ei * si;
    Wim[idx] = er * si + ei * sr;
}

// ---------------------------------------------------------------------------
// Kernel 5: complex GEMM  C = A @ B  (planar fp32 inputs), M=N=K=DIM.
// Block = 256 threads (8 waves), block tile 128x64, K tile 32 in LDS.
// Staging uses GLOBAL_LOAD_ASYNC_TO_LDS (ASYNCcnt) when available.
// Each wave: 32x32 complex tile = 2x2 WMMA tiles, 4x v_wmma_f32_16x16x4_f32
// per (mi,ni,k4).  Used for squarings (cstride=1, planar out) and the batched
// apply (cstride=2, interleaved complex64 out).
// ---------------------------------------------------------------------------
__launch_bounds__(256)
__global__ void qod_cgemm(const float* __restrict__ Ar, const float* __restrict__ Ai,
                          const float* __restrict__ Br0, const float* __restrict__ Bi0,
                          float* __restrict__ Cr0, float* __restrict__ Ci0,
                          long long bstrideB, long long bstrideC, int cstride) {
    __shared__ float lAre[BM * BKP];
    __shared__ float lAim[BM * BKP];
    __shared__ float lBre[BN * BKP];   // transposed: [n][k]
    __shared__ float lBim[BN * BKP];

    const int b = blockIdx.z;
    const float* Br = Br0 + (long long)b * bstrideB;
    const float* Bi = Bi0 + (long long)b * bstrideB;
    float* Cr = Cr0 + (long long)b * bstrideC;
    float* Ci = Ci0 + (long long)b * bstrideC;

    const int m0 = blockIdx.y * BM;
    const int n0 = blockIdx.x * BN;
    const int tid = threadIdx.x;
    const int wave = tid >> 5, lane = tid & 31;
    const int wm = (wave & 3) * 32;      // 4 waves along M
    const int wn = (wave >> 2) * 32;     // 2 waves along N
    const int half = lane >> 4;          // ISA f32 WMMA half-wave split
    const int l16 = lane & 15;

    v8f accRe[2][2], accIm[2][2];
#pragma unroll
    for (int mi = 0; mi < 2; ++mi)
#pragma unroll
        for (int ni = 0; ni < 2; ++ni) {
            accRe[mi][ni] = (v8f){0.f, 0.f, 0.f, 0.f, 0.f, 0.f, 0.f, 0.f};
            accIm[mi][ni] = (v8f){0.f, 0.f, 0.f, 0.f, 0.f, 0.f, 0.f, 0.f};
        }

    for (int kt = 0; kt < DIM; kt += BK) {
        __syncthreads();                  // previous tile fully consumed
#if USE_ASYNC_LDS
        // ---- async stage A tile (BMxBK): b64 per lane, LDS written by DMA ----
#pragma unroll
        for (int i = 0; i < 8; ++i) {
            int e = tid + i * 256;            // float2 index 0..2047
            int r = e >> 4;                   // row 0..127
            int c2 = (e & 15) << 1;           // even col 0..30
            size_t g = (size_t)(m0 + r) * DIM + kt + c2;
            __builtin_amdgcn_global_load_async_to_lds_b64(
                ASG64(Ar + g), ASL64(&lAre[r * BKP + c2]), 0, 0);
            __builtin_amdgcn_global_load_async_to_lds_b64(
                ASG64(Ai + g), ASL64(&lAim[r * BKP + c2]), 0, 0);
        }
        // ---- async stage B tile (BKxBN) transposed into [n][k]: b32 ----
#pragma unroll
        for (int i = 0; i < 8; ++i) {
            int e = tid + i * 256;            // 0..2047
            int k = e >> 6;                   // 0..31
            int n = e & 63;                   // 0..63
            size_t g = (size_t)(kt + k) * DIM + n0 + n;
            __builtin_amdgcn_global_load_async_to_lds_b32(
                ASG32(Br + g), ASL32(&lBre[n * BKP + k]), 0, 0);
            __builtin_amdgcn_global_load_async_to_lds_b32(
                ASG32(Bi + g), ASL32(&lBim[n * BKP + k]), 0, 0);
        }
        __builtin_amdgcn_s_wait_asynccnt(0);
#else
        // ---- stage A tile (BMxBK) as float2 via VGPRs ----
#pragma unroll
        for (int i = 0; i < 8; ++i) {
            int e = tid + i * 256;            // float2 index 0..2047
            int r = e >> 4;                   // row 0..127
            int c2 = (e & 15) << 1;           // even col 0..30
            size_t g = (size_t)(m0 + r) * DIM + kt + c2;
            float2 vr = *(const float2*)(Ar + g);
            float2 vi = *(const float2*)(Ai + g);
            *(float2*)&lAre[r * BKP + c2] = vr;
            *(float2*)&lAim[r * BKP + c2] = vi;
        }
        // ---- stage B tile (BKxBN) transposed into [n][k] ----
#pragma unroll
        for (int i = 0; i < 8; ++i) {
            int e = tid + i * 256;            // 0..2047
            int k = e >> 6;                   // 0..31
            int n = e & 63;                   // 0..63
            size_t g = (size_t)(kt + k) * DIM + n0 + n;
            lBre[n * BKP + k] = Br[g];
            lBim[n * BKP + k] = Bi[g];
        }
#endif
        __syncthreads();

        // prefetch next K tile (global_prefetch_b8); uniform branch, EXEC intact
        if (kt + BK < DIM) {
            if (tid < 128) {
                size_t g = (size_t)(m0 + tid) * DIM + kt + BK;
                __builtin_prefetch(Ar + g, 0, 3);
                __builtin_prefetch(Ai + g, 0, 3);
            } else if (tid < 160) {
                size_t g = (size_t)(kt + BK + (tid - 128)) * DIM + n0;
                __builtin_prefetch(Br + g, 0, 3);
                __builtin_prefetch(Bi + g, 0, 3);
            }
        }

#pragma unroll
        for (int k0 = 0; k0 < BK; k0 += 4) {
            const int kk = k0 + half * 2;     // lanes 0-15: K=k0,k0+1 ; 16-31: K=k0+2,k0+3
            v2f aRe[2], aIm[2], aImN[2], bRe[2], bIm[2];
#pragma unroll
            for (int mi = 0; mi < 2; ++mi) {
                int row = wm + mi * 16 + l16;
                aRe[mi] = *(const v2f*)&lAre[row * BKP + kk];
                aIm[mi] = *(const v2f*)&lAim[row * BKP + kk];
                aImN[mi] = -aIm[mi];          // f32 WMMA has no A-neg modifier
            }
#pragma unroll
            for (int ni = 0; ni < 2; ++ni) {
                int col = wn + ni * 16 + l16;
                bRe[ni] = *(const v2f*)&lBre[col * BKP + kk];
                bIm[ni] = *(const v2f*)&lBim[col * BKP + kk];
            }
#pragma unroll
            for (int mi = 0; mi < 2; ++mi)
#pragma unroll
                for (int ni = 0; ni < 2; ++ni) {
                    accRe[mi][ni] = __builtin_amdgcn_wmma_f32_16x16x4_f32(
                        false, aRe[mi], false, bRe[ni], (short)0, accRe[mi][ni], false, false);
                    accRe[mi][ni] = __builtin_amdgcn_wmma_f32_16x16x4_f32(
                        false, aImN[mi], false, bIm[ni], (short)0, accRe[mi][ni], false, false);
                    accIm[mi][ni] = __builtin_amdgcn_wmma_f32_16x16x4_f32(
                        false, aRe[mi], false, bIm[ni], (short)0, accIm[mi][ni], false, false);
                    accIm[mi][ni] = __builtin_amdgcn_wmma_f32_16x16x4_f32(
                        false, aIm[mi], false, bRe[ni], (short)0, accIm[mi][ni], false, false);
                }
        }
    }

    // ---- store: C/D layout: VGPR r -> M = r (lanes 0-15) / 8+r (lanes 16-31) ----
#pragma unroll
    for (int mi = 0; mi < 2; ++mi)
#pragma unroll
        for (int ni = 0; ni < 2; ++ni)
#pragma unroll
            for (int r = 0; r < 8; ++r) {
                int row = m0 + wm + mi * 16 + half * 8 + r;
                int col = n0 + wn + ni * 16 + l16;
                size_t e = (size_t)row * DIM + col;
                Cr[e * cstride] = accRe[mi][ni][r];
                Ci[e * cstride] = accIm[mi][ni][r];
            }
}

// ---------------------------------------------------------------------------
// Host orchestration (all on `stream`, graph-capture safe, deterministic).
// ---------------------------------------------------------------------------
extern "C" void kernel_launch(void* const* d_in, const int* in_sizes, int n_in,
                              void* d_out, int out_size, void* d_ws, size_t ws_size,
                              hipStream_t stream) {
    const float* in_re = (const float*)d_in[0];
    const float* in_im = (const float*)d_in[1];
    const float* th1   = (const float*)d_in[2];
    const float* th2   = (const float*)d_in[3];
    // d_in[4] = r : squeeze generator is identically zero -> S = I (unused)
    const float* bx    = (const float*)d_in[5];
    const float* bp    = (const float*)d_in[6];
    const float* kap   = (const float*)d_in[7];

    float* ws = (float*)d_ws;
    const size_t PL = (size_t)DIM * DIM;        // one plane = 1 MB
    float* Ere = ws + 0 * PL; float* Eim = ws + 1 * PL;
    float* Tre = ws + 2 * PL; float* Tim = ws + 3 * PL;
    float* Pre = ws + 4 * PL; float* Pim = ws + 5 * PL;
    float* Qre = ws + 6 * PL; float* Qim = ws + 7 * PL;
    float2* msup = (float2*)(ws + 8 * PL);
    float2* msub = msup + DIM;
    float2* dR   = msub + DIM;
    float2* dK   = dR + DIM;

    // 1) tables (alpha-scaled tridiagonal entries, rotation/Kerr diagonals)
    qod_tables<<<1, DIM, 0, stream>>>(th1, th2, bx, bp, kap, msup, msub, dR, dK);

    // 2) Taylor of exp(M/2^8): E = sum_{k<=8} Ms^k/k!  (tridiagonal products)
    qod_taylor_init<<<(int)(PL / 256), 256, 0, stream>>>(Ere, Eim, Pre, Pim, msup, msub);
    {
        float* pr = Pre; float* pi_ = Pim; float* qr = Qre; float* qi = Qim;
        for (int k = 2; k <= TAYLOR_ORDER; ++k) {
            qod_taylor_step<<<(int)(PL / 256), 256, 0, stream>>>(
                Ere, Eim, pr, pi_, qr, qi, msup, msub, 1.0f / (float)k);
            float* t;
            t = pr; pr = qr; qr = t;
            t = pi_; pi_ = qi; qi = t;
        }
    }

    // 3) square 8 times: E <- E@E  (complex WMMA GEMM, planar)
    float* ar = Ere; float* ai = Eim; float* cr = Tre; float* ci = Tim;
    {
        dim3 grid(DIM / BN, DIM / BM, 1);
        for (int s = 0; s < SQUARINGS; ++s) {
            qod_cgemm<<<grid, 256, 0, stream>>>(ar, ai, ar, ai, cr, ci, 0LL, 0LL, 1);
            float* t;
            t = ar; ar = cr; cr = t;
            t = ai; ai = ci; ci = t;
        }
    }

    // 4) fuse: W = diag_K * D * diag_R  (into the now-free P planes)
    qod_fuse<<<(int)(PL / 256), 256, 0, stream>>>(ar, ai, dK, dR, Pre, Pim);

    // 5) batched apply: Y[b] = W @ X[b], output interleaved complex64
    {
        dim3 grid(DIM / BN, DIM / BM, BATCH);
        float* out = (float*)d_out;
        qod_cgemm<<<grid, 256, 0, stream>>>(Pre, Pim, in_re, in_im,
                                            out, out + 1,
                                            (long long)PL, (long long)(PL * 2), 2);
    }
}